// ParallelTransport_39264591020517
// MI455X (gfx1250) — compile-verified
//
#include <hip/hip_runtime.h>

// ParallelTransport: out[b,e,c,i] = R(rho[e])[i,j] * x[b, row[e], c, j]
// R = [[cos, -sin], [sin, cos]],  B=1, C=32, 2-vectors per channel.
//
// Memory-bound kernel (AI ~ 0.4 flop/byte, HBM floor ~430 MB / 23.3 TB/s).
// MI455X strategy:
//  - 8 threads per edge, 2x b128 per thread -> fully coalesced 256B per edge.
//  - Output (410 MB write-once stream): TH_STORE_NT so it does not evict the
//    12.8 MB x[] working set from the 192 MB L2.
//  - edge_index/rho streamed once -> TH_LOAD_NT (GVS + scale_offset form).
//  - x[] gathers regular-temporal (L2-resident, ~32x average reuse).
//  - __sincosf -> v_sin_f32/v_cos_f32 trans ops (co-execute with VMEM).
//  - Single 32-bit byte offsets per access (max 409.6 MB < 4 GB) so every
//    address lowers to SGPR-base + 32-bit VGPR offset, no 64-bit VALU adds.

typedef __attribute__((ext_vector_type(4))) float v4f;  // native vector for NT builtins

__global__ __launch_bounds__(256) void ParallelTransport_39264591020517_kernel(
    const float* __restrict__ x,        // (1, N, 32, 2) -> 256 B per node
    const int*   __restrict__ row_idx,  // edge_index[0], E entries
    const float* __restrict__ rho,      // E entries
    float*       __restrict__ out,      // (1, E, 32, 2) -> 256 B per edge
    unsigned E)
{
    const unsigned tid = blockIdx.x * 256u + threadIdx.x;  // < 12.8M, fits u32
    const unsigned e   = tid >> 3;                         // edge id
    if (e >= E) return;
    const unsigned p16 = (tid & 7u) * 16u;  // byte offset of this thread's 16B slice

    // Streamed-once scalars: non-temporal loads (don't pollute L2).
    const unsigned row = (unsigned)__builtin_nontemporal_load(row_idx + e);
    const float    r   = __builtin_nontemporal_load(rho + e);

    float s, c;
    __sincosf(r, &s, &c);

    // Gather: single 32-bit byte offset -> saddr + voffset addressing.
    // x[] is 12.8 MB total -> L2 hits; 256B contiguous per edge.
    const v4f* __restrict__ src =
        (const v4f*)((const char*)x + (row * 256u + p16));
    const v4f v0 = src[0];   // bytes [p16, p16+16)      : channels 2p, 2p+1
    const v4f v1 = src[8];   // bytes [128+p16, ...)     : channels 2p+16, 2p+17

    // Per-channel 2-vector rotation: (x0,x1) -> (c*x0 - s*x1, s*x0 + c*x1)
    v4f o0, o1;
    o0.x = fmaf(c, v0.x, -s * v0.y);  o0.y = fmaf(s, v0.x, c * v0.y);
    o0.z = fmaf(c, v0.z, -s * v0.w);  o0.w = fmaf(s, v0.z, c * v0.w);
    o1.x = fmaf(c, v1.x, -s * v1.y);  o1.y = fmaf(s, v1.x, c * v1.y);
    o1.z = fmaf(c, v1.z, -s * v1.w);  o1.w = fmaf(s, v1.z, c * v1.w);

    // Write-once stream: NT stores keep L2 for the x[] gather set.
    v4f* dst = (v4f*)((char*)out + (e * 256u + p16));
    __builtin_nontemporal_store(o0, dst);
    __builtin_nontemporal_store(o1, dst + 8);
}

extern "C" void kernel_launch(void* const* d_in, const int* in_sizes, int n_in,
                              void* d_out, int out_size, void* d_ws, size_t ws_size,
                              hipStream_t stream)
{
    const float* x    = (const float*)d_in[0];
    const int*   edge = (const int*)d_in[1];   // (2, E); row = first E entries
    const float* rho  = (const float*)d_in[2];
    float*       out  = (float*)d_out;

    const unsigned E = (unsigned)in_sizes[2];  // rho has exactly E elements

    const long long total_threads = (long long)E * 8;
    const int blocks = (int)((total_threads + 255) / 256);

    hipLaunchKernelGGL(ParallelTransport_39264591020517_kernel,
                       dim3(blocks), dim3(256), 0, stream,
                       x, edge, rho, out, E);
}